// TASI_GNN_2388001816922
// MI455X (gfx1250) — compile-verified
//
#include <hip/hip_runtime.h>
#include <math.h>

// ---------------------------------------------------------------------------
// TASI-GNN forward for MI455X (gfx1250, wave32).
// All GEMMs use V_WMMA_F32_16X16X4_F32.  Main K-loops are branch-free:
// out-of-range rows/columns are handled by pointer clamping (their outputs
// are never stored), and only the K-tail step uses select-zeroing.
// ---------------------------------------------------------------------------

typedef __attribute__((ext_vector_type(2))) float v2f;
typedef __attribute__((ext_vector_type(8))) float v8f;

#define EDIM   128
#define BSZ    512
#define LSEQ   50
#define LP1    51
#define ITEMS  50000
#define NEGVAL -1000000000.0f
#define GAMMA_C 0.5f
#define THETA_C 12.0f
#define OMEGA_C 0.1f
#define SELU_S  1.0507009873554805f
#define SELU_A  1.6732632423543772f

// ---------------- workspace layout (float offsets) -------------------------
static const long F_WITEM = 0;                            // 50001*128
static const long F_POSR  = F_WITEM + 50001L * 128;       // 50*128
static const long F_X     = F_POSR  + 50L * 128;          // 25600*256
static const long F_TMP1  = F_X     + 25600L * 256;       // 25600*256
static const long F_TMP2  = F_TMP1  + 25600L * 256;       // 25600*256
static const long F_ADJ   = F_TMP2  + 25600L * 256;       // 25600*512
static const long F_XG    = F_ADJ   + 25600L * 512;       // 25600*256
static const long F_END   = F_XG    + 25600L * 256;       // ~45.7M floats (~183MB)
// phase 2 overlays the (now dead) X/TMP1/TMP2/ADJ region [F_X, F_XG)
static const long F_XTP   = F_X;                          // 26112*256
static const long F_Q     = F_XTP + 26112L * 256;         // 26112*256 (later CHAT)
static const long F_SC    = F_Q   + 26112L * 256;         // 26112*51
static const long F_AW    = F_SC  + 26112L * 51;          // 26112*51
static const long F_ATT   = F_AW  + 26112L * 51;          // 26112*256 (later SG)
static const long F_F1    = F_ATT + 26112L * 256;         // 26112*256
static const long F_SM    = F_F1  + 26112L * 256;         // smalls (< F_XG)
static const long F_ALPHA1 = F_SM;                  // 512
static const long F_TGT    = F_ALPHA1 + 512;        // 512*256
static const long F_CLAST  = F_TGT   + 131072;      // 512*256
static const long F_SIMV   = F_CLAST + 131072;      // 512
static const long F_SIMM   = F_SIMV  + 512;         // 8
static const long F_CAT1   = F_SIMM  + 8;           // 512*512
static const long F_POS    = F_CAT1  + 262144;      // 512*256
static const long F_ALPHA2 = F_POS   + 131072;      // 512
static const long F_TG     = F_ALPHA2+ 512;         // 512*256
static const long F_SROW   = F_TG    + 131072;      // 25600
static const long F_WROW   = F_SROW  + 25600;       // 25600
static const long F_CGLOB  = F_WROW  + 25600;       // 512*256
static const long F_CAT2   = F_CGLOB + 131072;      // 512*512
static const long F_SH     = F_CAT2  + 262144;      // 512*128
static const long F_HN     = F_SH    + 65536;       // 512*128
static const long F_SIM2   = F_HN    + 65536;       // 512*512
static const long F_BETA   = F_SIM2  + 262144;      // 512*10
static const long F_IDX    = F_BETA  + 5120;        // 512*10 (ints)
static const long F_SFIN   = F_IDX   + 5120;        // 512*128
static const long F_INORM  = F_XTP;                 // 50000*128 (XTP dead by then)

// ---------------------------- helpers ---------------------------------------
__device__ __forceinline__ float sigmoidf(float x) { return 1.f / (1.f + expf(-x)); }

__device__ __forceinline__ float waveSum(float v) {
  #pragma unroll
  for (int o = 16; o > 0; o >>= 1) v += __shfl_xor(v, o, 32);
  return v;
}
__device__ __forceinline__ float waveMax(float v) {
  #pragma unroll
  for (int o = 16; o > 0; o >>= 1) v = fmaxf(v, __shfl_xor(v, o, 32));
  return v;
}

template <int NT>
__device__ __forceinline__ float blockSum(float v) {
  __shared__ float s[NT / 32];
  __shared__ float r;
  v = waveSum(v);
  int w = threadIdx.x >> 5;
  if ((threadIdx.x & 31) == 0) s[w] = v;
  __syncthreads();
  if (w == 0) {
    float x = (threadIdx.x < NT / 32) ? s[threadIdx.x] : 0.f;
    x = waveSum(x);
    if (threadIdx.x == 0) r = x;
  }
  __syncthreads();
  float out = r;
  __syncthreads();
  return out;
}

__device__ __forceinline__ v8f wmma4(v2f a, v2f b, v8f c) {
  // D = A(16x4,f32) * B(4x16,f32) + C(16x16,f32)   [V_WMMA_F32_16X16X4_F32]
  return __builtin_amdgcn_wmma_f32_16x16x4_f32(false, a, false, b, (short)0, c,
                                               false, false);
}

// ------------------------- generic WMMA GEMM --------------------------------
// out[m,n] = act( scale * sum_k A[m,k] * (transB ? B[n,k] : B[k,n])
//                 + bias[n] + rowAdd[(m/rowDiv)*N + n] )
// One wave computes a 16 x (16*NTJ) output strip: a single A fragment feeds
// NTJ WMMAs per K-step.  All main-loop loads are unconditional: out-of-range
// rows/cols use clamped pointers (D rows/cols are independent, and those
// outputs are discarded by the guarded store).  Only the K-tail step zero-
// selects, so EXEC stays all-ones and the inner loop is branch-free.
template <int NTJ>
__global__ void wmma_gemm(const float* __restrict__ A, const float* __restrict__ Bm,
                          const float* __restrict__ bias,
                          const float* __restrict__ rowAdd, int rowDiv, long rowAddSb,
                          float* __restrict__ out,
                          int M, int N, int K,
                          long sAb, long sBb, long sOb,
                          int lda, int ldb, int ldo,
                          int transB, float scale, int act) {
  int bz = blockIdx.z;
  A  += (long)bz * sAb;
  Bm += (long)bz * sBb;
  out += (long)bz * sOb;
  const float* rA = rowAdd ? rowAdd + (long)bz * rowAddSb : nullptr;

  int n0 = blockIdx.x * (16 * NTJ), m0 = blockIdx.y * 16;
  int lane = threadIdx.x & 31;
  int half = lane >> 4;      // K pair selector
  int mn = lane & 15;
  int am = min(m0 + mn, M - 1);                 // clamped A row
  const float* Arow = A + (long)am * lda;

  const float* Brow[NTJ];                       // clamped B rows (transB)
  int bnc[NTJ];                                 // clamped B cols (NN)
  #pragma unroll
  for (int j = 0; j < NTJ; ++j) {
    int bn = min(n0 + 16 * j + mn, N - 1);
    Brow[j] = Bm + (long)bn * ldb;
    bnc[j] = bn;
  }

  v8f acc[NTJ];
  #pragma unroll
  for (int j = 0; j < NTJ; ++j) acc[j] = (v8f){0.f,0.f,0.f,0.f,0.f,0.f,0.f,0.f};

  int K4 = K & ~3;
  for (int k0 = 0; k0 < K4; k0 += 4) {
    int ks = k0 + 2 * half;
    v2f a;  a.x = Arow[ks];  a.y = Arow[ks + 1];          // merges to b64
    #pragma unroll
    for (int j = 0; j < NTJ; ++j) {
      v2f b;
      if (transB) { b.x = Brow[j][ks];  b.y = Brow[j][ks + 1]; }
      else        { b.x = Bm[(long)ks * ldb + bnc[j]];
                    b.y = Bm[(long)(ks + 1) * ldb + bnc[j]]; }
      acc[j] = wmma4(a, b, acc[j]);
    }
  }
  if (K4 < K) {                                  // guarded K tail (once)
    int ks = K4 + 2 * half;
    int c0 = min(ks, K - 1), c1 = min(ks + 1, K - 1);
    float ax = Arow[c0], ay = Arow[c1];
    v2f a;  a.x = (ks < K) ? ax : 0.f;  a.y = (ks + 1 < K) ? ay : 0.f;
    #pragma unroll
    for (int j = 0; j < NTJ; ++j) {
      float bx, by;
      if (transB) { bx = Brow[j][c0];  by = Brow[j][c1]; }
      else        { bx = Bm[(long)c0 * ldb + bnc[j]];
                    by = Bm[(long)c1 * ldb + bnc[j]]; }
      v2f b;  b.x = (ks < K) ? bx : 0.f;  b.y = (ks + 1 < K) ? by : 0.f;
      acc[j] = wmma4(a, b, acc[j]);
    }
  }

  #pragma unroll
  for (int j = 0; j < NTJ; ++j) {
    int n = n0 + 16 * j + mn;
    if (n >= N) continue;
    #pragma unroll
    for (int i = 0; i < 8; ++i) {
      int m = m0 + i + 8 * half;                 // D layout: M = i + 8*(lane>=16)
      if (m >= M) continue;
      float v = acc[j][i] * scale;
      if (bias) v += bias[n];
      if (rA)   v += rA[(long)(m / rowDiv) * N + n];
      if (act == 1) v = fmaxf(v, 0.f);
      else if (act == 2) v = v > 0.f ? SELU_S * v : SELU_S * SELU_A * (expf(v) - 1.f);
      out[(long)m * ldo + n] = v;
    }
  }
}

// ------------------- fused GRU (gi/gh never materialized) -------------------
// One wave per 16x16 hidden tile accumulates six WMMA tiles (ir,iz,in over
// K=512 of adjcat; hr,hz,hn over K=256 of X) and applies the gate math
// in-register.  6 acc * 8 VGPR = 48 VGPRs; loop is branch-free.
__global__ void gru_fused(const float* __restrict__ adj, const float* __restrict__ X,
                          const float* __restrict__ Wih, const float* __restrict__ bih,
                          const float* __restrict__ Whh, const float* __restrict__ bhh,
                          float* __restrict__ Xg) {
  int n0 = blockIdx.x * 16, m0 = blockIdx.y * 16;
  int lane = threadIdx.x & 31, half = lane >> 4, mn = lane & 15;
  int am = m0 + mn, bn = n0 + mn;
  v8f ir = {0,0,0,0,0,0,0,0}, iz = ir, inn = ir, hr = ir, hz = ir, hnn = ir;

  const float* arow = adj + (long)am * 512;
  const float* w0 = Wih + (long)bn * 512;
  const float* w1 = Wih + (long)(bn + 256) * 512;
  const float* w2 = Wih + (long)(bn + 512) * 512;
  for (int k0 = 0; k0 < 512; k0 += 4) {
    int ks = k0 + 2 * half;
    v2f a;  a.x = arow[ks];  a.y = arow[ks + 1];
    v2f b0, b1, b2;
    b0.x = w0[ks];  b0.y = w0[ks + 1];
    b1.x = w1[ks];  b1.y = w1[ks + 1];
    b2.x = w2[ks];  b2.y = w2[ks + 1];
    ir  = wmma4(a, b0, ir);
    iz  = wmma4(a, b1, iz);
    inn = wmma4(a, b2, inn);
  }
  const float* xrow = X + (long)am * 256;
  const float* u0 = Whh + (long)bn * 256;
  const float* u1 = Whh + (long)(bn + 256) * 256;
  const float* u2 = Whh + (long)(bn + 512) * 256;
  for (int k0 = 0; k0 < 256; k0 += 4) {
    int ks = k0 + 2 * half;
    v2f a;  a.x = xrow[ks];  a.y = xrow[ks + 1];
    v2f b0, b1, b2;
    b0.x = u0[ks];  b0.y = u0[ks + 1];
    b1.x = u1[ks];  b1.y = u1[ks + 1];
    b2.x = u2[ks];  b2.y = u2[ks + 1];
    hr  = wmma4(a, b0, hr);
    hz  = wmma4(a, b1, hz);
    hnn = wmma4(a, b2, hnn);
  }
  int n = n0 + mn;
  #pragma unroll
  for (int i = 0; i < 8; ++i) {
    int m = m0 + i + 8 * half;
    float r  = sigmoidf(ir[i] + bih[n]       + hr[i]  + bhh[n]);
    float z  = sigmoidf(iz[i] + bih[256 + n] + hz[i]  + bhh[256 + n]);
    float ng = tanhf(inn[i] + bih[512 + n] + r * (hnn[i] + bhh[512 + n]));
    float h  = X[(long)m * 256 + n];
    Xg[(long)m * 256 + n] = (1.f - z) * ng + z * h;
  }
}

// --------------------------- small kernels ----------------------------------
__global__ void renorm_rows(const float* __restrict__ in, float* __restrict__ out) {
  long row = blockIdx.x;
  int t = threadIdx.x;                         // 128 threads, dim 128
  float x = in[row * EDIM + t];
  float ss = blockSum<128>(x * x);
  float n = sqrtf(ss);
  float s = n > 1.5f ? 1.5f / (n + 1e-7f) : 1.f;
  out[row * EDIM + t] = x * s;
}

__global__ void l2norm_rows(const float* __restrict__ in, float* __restrict__ out,
                            float minv) {
  long row = blockIdx.x;
  int t = threadIdx.x;                         // 128 threads, dim 128
  float x = in[row * EDIM + t];
  float ss = blockSum<128>(x * x);
  out[row * EDIM + t] = x / fmaxf(sqrtf(ss), minv);
}

__global__ void build_X(const float* __restrict__ witem, const float* __restrict__ posr,
                        const int* __restrict__ item, float* __restrict__ X) {
  int row = blockIdx.x;                        // 25600 rows, 128 threads
  int l = row % LSEQ, t = threadIdx.x;
  long it = item[row];
  X[(long)row * 256 + t]        = witem[it * EDIM + t];
  X[(long)row * 256 + 128 + t]  = posr[l * EDIM + t];
}

__global__ void build_Xtp(const float* __restrict__ Xg, const int* __restrict__ alias,
                          float* __restrict__ Xtp) {
  int row = blockIdx.x;                        // 26112 rows, 256 threads
  int b = row / LP1, i = row % LP1, t = threadIdx.x;
  float v = 0.f;
  if (i < LSEQ) {
    int a = alias[b * LSEQ + i];
    v = Xg[((long)b * LSEQ + a) * 256 + t];
  }
  Xtp[(long)row * 256 + t] = v;
}

__global__ void alpha_vec(const float* __restrict__ src, long strideB,
                          const float* __restrict__ Wa, const float* __restrict__ ba,
                          float* __restrict__ out) {
  int b = blockIdx.x;                          // one wave per batch row
  const float* x = src + (long)b * strideB;
  float d = 0.f;
  for (int j = threadIdx.x; j < 256; j += 32) d += x[j] * Wa[j];
  d = waveSum(d);
  if (threadIdx.x == 0) out[b] = fmaxf(sigmoidf(d + ba[0]) + 1.f, 1.f + 1e-5f);
}

// entmax_bisect over a row of length d<=64, alpha per (row/alphaDiv).
__global__ void entmax_rows(const float* __restrict__ x, const float* __restrict__ al,
                            int alphaDiv, float* __restrict__ out, int d) {
  int row = blockIdx.x, lane = threadIdx.x;
  float alpha = al[row / alphaDiv];
  float am1 = alpha - 1.f, inv = 1.f / am1;
  const float* xr = x + (long)row * d;
  float xs0 = (lane < d)      ? xr[lane] * am1       : -3.0e38f;
  float xs1 = (lane + 32 < d) ? xr[lane + 32] * am1  : -3.0e38f;
  float mx = waveMax(fmaxf(xs0, xs1));
  float tau = mx - 1.f;
  float tau_hi = mx - powf(1.f / (float)d, am1);
  auto pfn = [&](float v) { return v > 0.f ? powf(v, inv) : 0.f; };
  float f_lo = waveSum(pfn(xs0 - tau) + pfn(xs1 - tau)) - 1.f;
  float dm = tau_hi - tau;
  for (int it = 0; it < 50; ++it) {
    dm *= 0.5f;
    float tm = tau + dm;
    float fm = waveSum(pfn(xs0 - tm) + pfn(xs1 - tm)) - 1.f;
    if (fm * f_lo >= 0.f) tau = tm;
  }
  float p0 = pfn(xs0 - tau), p1 = pfn(xs1 - tau);
  float s = 1.f / waveSum(p0 + p1);
  if (lane < d)      out[(long)row * d + lane]      = p0 * s;
  if (lane + 32 < d) out[(long)row * d + lane + 32] = p1 * s;
}

__global__ void layer_norm_rows(float* __restrict__ buf, const float* __restrict__ g,
                                const float* __restrict__ bb) {
  long row = blockIdx.x;                       // 256 threads, dim 256
  int t = threadIdx.x;
  float v = buf[row * 256 + t];
  float mu = blockSum<256>(v) * (1.f / 256.f);
  float dv = v - mu;
  float var = blockSum<256>(dv * dv) * (1.f / 256.f);
  buf[row * 256 + t] = dv * rsqrtf(var + 1e-12f) * g[t] + bb[t];
}

__global__ void post_chat(const float* __restrict__ chat, const int* __restrict__ mask,
                          float* __restrict__ tgt, float* __restrict__ clast,
                          float* __restrict__ cat1, float* __restrict__ simv) {
  int b = blockIdx.x, t = threadIdx.x;         // 256 threads
  int sl = 0;
  for (int l = 0; l < LSEQ; ++l) sl += mask[b * LSEQ + l];
  sl -= 1;
  float tg = chat[((long)b * LP1 + LSEQ) * 256 + t];
  float cl = chat[((long)b * LP1 + sl) * 256 + t];
  tgt[(long)b * 256 + t] = tg;
  clast[(long)b * 256 + t] = cl;
  cat1[(long)b * 512 + t] = tg;
  cat1[(long)b * 512 + 256 + t] = cl;
  float dot = blockSum<256>(tg * cl);
  float na  = blockSum<256>(tg * tg);
  float nb  = blockSum<256>(cl * cl);
  if (t == 0)
    simv[b] = dot / (fmaxf(sqrtf(na), 1e-8f) * fmaxf(sqrtf(nb), 1e-8f));
}

__global__ void mean512(const float* __restrict__ v, float* __restrict__ m) {
  float s = blockSum<512>(v[threadIdx.x]);
  if (threadIdx.x == 0) m[0] = s * (1.f / 512.f);
}

__global__ void select_target(const float* __restrict__ simv, const float* __restrict__ mn,
                              const float* __restrict__ pos, float* __restrict__ tgt) {
  int b = blockIdx.x, t = threadIdx.x;
  bool c = (simv[b] - GAMMA_C * mn[0]) > 0.f;
  long i = (long)b * 256 + t;
  tgt[i] = c ? pos[i] : tgt[i];
}

__global__ void compute_s(const float* __restrict__ sg, const float* __restrict__ wg0,
                          const int* __restrict__ mask, float* __restrict__ s) {
  int row = blockIdx.x;                        // 25600 rows, one wave each
  const float* x = sg + (long)row * 256;
  float d = 0.f;
  for (int j = threadIdx.x; j < 256; j += 32) d += x[j] * wg0[j];
  d = waveSum(d);
  if (threadIdx.x == 0) s[row] = (mask[row] != 0) ? d : NEGVAL;
}

__global__ void c_global(const float* __restrict__ w, const float* __restrict__ xtp,
                         float* __restrict__ cg) {
  int b = blockIdx.x, t = threadIdx.x;         // 256 threads
  float acc = 0.f;
  for (int l = 0; l < LSEQ; ++l)
    acc += w[b * LSEQ + l] * xtp[((long)b * LP1 + l) * 256 + t];
  cg[(long)b * 256 + t] = acc;
}

__global__ void build_cat2(const float* __restrict__ cg, const float* __restrict__ tgt,
                           float* __restrict__ cat2) {
  int b = blockIdx.x, t = threadIdx.x;
  cat2[(long)b * 512 + t] = cg[(long)b * 256 + t];
  cat2[(long)b * 512 + 256 + t] = tgt[(long)b * 256 + t];
}

__global__ void topk_beta(const float* __restrict__ sim2, float* __restrict__ beta,
                          int* __restrict__ idx) {
  int b = blockIdx.x, lane = threadIdx.x;      // one wave, row of 512
  float v[16];
  #pragma unroll
  for (int j = 0; j < 16; ++j) v[j] = sim2[(long)b * 512 + j * 32 + lane];
  float tv[10];
  int   ti[10];
  for (int t = 0; t < 10; ++t) {
    float bv = -3.0e38f;
    int bi = 0x7fffffff;
    #pragma unroll
    for (int j = 0; j < 16; ++j)
      if (v[j] > bv) { bv = v[j]; bi = j * 32 + lane; }
    #pragma unroll
    for (int o = 16; o > 0; o >>= 1) {
      float ov = __shfl_xor(bv, o, 32);
      int   oi = __shfl_xor(bi, o, 32);
      if (ov > bv || (ov == bv && oi < bi)) { bv = ov; bi = oi; }
    }
    if ((bi & 31) == lane) v[bi >> 5] = -3.0e38f;  // owner retires the winner
    tv[t] = bv;
    ti[t] = bi;
  }
  if (lane == 0) {
    float m = tv[0], sum = 0.f, e[10];
    for (int k = 0; k < 10; ++k) { e[k] = expf(THETA_C * (tv[k] - m)); sum += e[k]; }
    for (int k = 0; k < 10; ++k) {
      beta[b * 10 + k] = e[k] / sum;
      idx[b * 10 + k]  = ti[k];
    }
  }
}

__global__ void final_mix(const float* __restrict__ sh, const float* __restrict__ beta,
                          const int* __restrict__ idx, float* __restrict__ sfin) {
  int b = blockIdx.x, d = threadIdx.x;         // 128 threads
  float sv = sh[(long)b * EDIM + d];
  float ss = blockSum<128>(sv * sv);
  float nrm = fmaxf(sqrtf(ss), 1e-12f);
  float hs = 0.f;
  for (int k = 0; k < 10; ++k)
    hs += beta[b * 10 + k] * sh[(long)idx[b * 10 + k] * EDIM + d];
  sfin[(long)b * EDIM + d] = sv / nrm + OMEGA_C * hs;
}

// ------------------------------ launcher ------------------------------------
static inline void gemm(hipStream_t s, const float* A, const float* B, const float* bias,
                        const float* rowAdd, int rowDiv, long rowAddSb, float* out,
                        int M, int N, int K, long sAb, long sBb, long sOb,
                        int lda, int ldb, int ldo, int transB, float scale, int act,
                        int batch) {
  dim3 g((N + 63) / 64, (M + 15) / 16, batch);   // 16 x 64 strip per wave
  wmma_gemm<4><<<g, 32, 0, s>>>(A, B, bias, rowAdd, rowDiv, rowAddSb, out, M, N, K,
                                sAb, sBb, sOb, lda, ldb, ldo, transB, scale, act);
}

extern "C" void kernel_launch(void* const* d_in, const int* in_sizes, int n_in,
                              void* d_out, int out_size, void* d_ws, size_t ws_size,
                              hipStream_t stream) {
  (void)in_sizes; (void)n_in; (void)out_size; (void)ws_size;
  const float* item_emb = (const float*)d_in[0];
  const float* pos_emb  = (const float*)d_in[1];
  const float* W_Hin  = (const float*)d_in[2];
  const float* W_Hout = (const float*)d_in[3];
  const float* b_in   = (const float*)d_in[4];
  const float* b_out  = (const float*)d_in[5];
  const float* W_ih   = (const float*)d_in[6];
  const float* b_ih   = (const float*)d_in[7];
  const float* W_hh   = (const float*)d_in[8];
  const float* b_hh   = (const float*)d_in[9];
  const float* W_a    = (const float*)d_in[10];
  const float* b_a    = (const float*)d_in[11];
  const float* W_q    = (const float*)d_in[12];
  const float* b_q    = (const float*)d_in[13];
  const float* W_f1   = (const float*)d_in[14];
  const float* b_f1   = (const float*)d_in[15];
  const float* W_f2   = (const float*)d_in[16];
  const float* b_f2   = (const float*)d_in[17];
  const float* ln_g   = (const float*)d_in[18];
  const float* ln_b   = (const float*)d_in[19];
  const float* W_c    = (const float*)d_in[20];
  const float* W_g1   = (const float*)d_in[21];
  const float* W_g2   = (const float*)d_in[22];
  const float* W_g0   = (const float*)d_in[23];
  const float* b_al   = (const float*)d_in[24];
  const float* W_h    = (const float*)d_in[25];
  const float* b_h    = (const float*)d_in[26];
  const float* Amat   = (const float*)d_in[27];
  const int* aliasI   = (const int*)d_in[28];
  const int* itemI    = (const int*)d_in[29];
  const int* maskI    = (const int*)d_in[30];
  float* ws  = (float*)d_ws;
  float* out = (float*)d_out;

  // ---- embeddings / X ----
  renorm_rows<<<50001, 128, 0, stream>>>(item_emb, ws + F_WITEM);
  renorm_rows<<<50, 128, 0, stream>>>(pos_emb, ws + F_POSR);
  build_X<<<25600, 128, 0, stream>>>(ws + F_WITEM, ws + F_POSR, itemI, ws + F_X);

  // ---- GNN message passing ----
  gemm(stream, ws + F_X, W_Hin, nullptr, nullptr, 1, 0, ws + F_TMP1,
       25600, 256, 256, 0, 0, 0, 256, 256, 256, 1, 1.f, 0, 1);
  gemm(stream, ws + F_X, W_Hout, nullptr, nullptr, 1, 0, ws + F_TMP2,
       25600, 256, 256, 0, 0, 0, 256, 256, 256, 1, 1.f, 0, 1);
  gemm(stream, Amat, ws + F_TMP1, b_in, nullptr, 1, 0, ws + F_ADJ,
       50, 256, 50, 5000, 12800, 25600, 100, 256, 512, 0, 1.f, 0, 512);
  gemm(stream, Amat + 50, ws + F_TMP2, b_out, nullptr, 1, 0, ws + F_ADJ + 256,
       50, 256, 50, 5000, 12800, 25600, 100, 256, 512, 0, 1.f, 0, 512);

  // ---- fused GRU cell ----
  gru_fused<<<dim3(16, 1600), 32, 0, stream>>>(ws + F_ADJ, ws + F_X, W_ih, b_ih,
                                               W_hh, b_hh, ws + F_XG);

  // ---- session gather + padded Xtp ----
  build_Xtp<<<26112, 256, 0, stream>>>(ws + F_XG, aliasI, ws + F_XTP);
  alpha_vec<<<512, 32, 0, stream>>>(ws + F_XTP + 50L * 256, 51L * 256, W_a, b_a,
                                    ws + F_ALPHA1);

  // ---- entmax attention block ----
  gemm(stream, ws + F_XTP, W_q, b_q, nullptr, 1, 0, ws + F_Q,
       26112, 256, 256, 0, 0, 0, 256, 256, 256, 1, 1.f, 1, 1);
  gemm(stream, ws + F_Q, ws + F_XTP, nullptr, nullptr, 1, 0, ws + F_SC,
       51, 51, 256, 13056, 13056, 2601, 256, 256, 51, 1, 1.f / 16.f, 0, 512);
  entmax_rows<<<26112, 32, 0, stream>>>(ws + F_SC, ws + F_ALPHA1, 51, ws + F_AW, 51);
  gemm(stream, ws + F_AW, ws + F_XTP, nullptr, nullptr, 1, 0, ws + F_ATT,
       51, 256, 51, 2601, 13056, 13056, 51, 256, 256, 0, 1.f, 0, 512);

  // ---- FFN + residual + LayerNorm (CHAT overlays dead Q) ----
  gemm(stream, ws + F_ATT, W_f1, b_f1, nullptr, 1, 0, ws + F_F1,
       26112, 256, 256, 0, 0, 0, 256, 256, 256, 1, 1.f, 1, 1);
  gemm(stream, ws + F_F1, W_f2, b_f2, ws + F_ATT, 1, 0, ws + F_Q,
       26112, 256, 256, 0, 0, 0, 256, 256, 256, 1, 1.f, 0, 1);
  layer_norm_rows<<<26112, 256, 0, stream>>>(ws + F_Q, ln_g, ln_b);

  // ---- target selection ----
  post_chat<<<512, 256, 0, stream>>>(ws + F_Q, maskI, ws + F_TGT, ws + F_CLAST,
                                     ws + F_CAT1, ws + F_SIMV);
  mean512<<<1, 512, 0, stream>>>(ws + F_SIMV, ws + F_SIMM);
  gemm(stream, ws + F_CAT1, W_c, nullptr, nullptr, 1, 0, ws + F_POS,
       512, 256, 512, 0, 0, 0, 512, 512, 256, 1, 1.f, 0, 1);
  select_target<<<512, 256, 0, stream>>>(ws + F_SIMV, ws + F_SIMM, ws + F_POS,
                                         ws + F_TGT);
  alpha_vec<<<512, 32, 0, stream>>>(ws + F_TGT, 256, W_a, b_a, ws + F_ALPHA2);

  // ---- global gate (SG overlays dead ATT) ----
  gemm(stream, ws + F_TGT, W_g2, nullptr, nullptr, 1, 0, ws + F_TG,
       512, 256, 256, 0, 0, 0, 256, 256, 256, 1, 1.f, 0, 1);
  gemm(stream, ws + F_Q, W_g1, b_al, ws + F_TG, 1 << 30, 256, ws + F_ATT,
       50, 256, 256, 13056, 0, 12800, 256, 256, 256, 1, 1.f, 1, 512);
  compute_s<<<25600, 32, 0, stream>>>(ws + F_ATT, W_g0, maskI, ws + F_SROW);
  entmax_rows<<<512, 32, 0, stream>>>(ws + F_SROW, ws + F_ALPHA2, 1, ws + F_WROW, 50);
  c_global<<<512, 256, 0, stream>>>(ws + F_WROW, ws + F_XTP, ws + F_CGLOB);

  // ---- SELU head + neighbor smoothing ----
  build_cat2<<<512, 256, 0, stream>>>(ws + F_CGLOB, ws + F_TGT, ws + F_CAT2);
  gemm(stream, ws + F_CAT2, W_h, b_h, nullptr, 1, 0, ws + F_SH,
       512, 128, 512, 0, 0, 0, 512, 512, 128, 1, 1.f, 2, 1);
  l2norm_rows<<<512, 128, 0, stream>>>(ws + F_SH, ws + F_HN, 1e-8f);
  gemm(stream, ws + F_HN, ws + F_HN, nullptr, nullptr, 1, 0, ws + F_SIM2,
       512, 512, 128, 0, 0, 0, 128, 128, 512, 1, 1.f, 0, 1);
  topk_beta<<<512, 32, 0, stream>>>(ws + F_SIM2, ws + F_BETA, (int*)(ws + F_IDX));
  final_mix<<<512, 128, 0, stream>>>(ws + F_SH, ws + F_BETA, (int*)(ws + F_IDX),
                                     ws + F_SFIN);

  // ---- final scoring: (512x128) x (50000x128)^T  (INORM overlays dead XTP) ----
  l2norm_rows<<<50000, 128, 0, stream>>>(ws + F_WITEM + EDIM, ws + F_INORM, 1e-12f);
  gemm(stream, ws + F_SFIN, ws + F_INORM, nullptr, nullptr, 1, 0, out,
       512, 50000, 128, 0, 0, 0, 128, 128, 50000, 1, 1.f, 0, 1);
}